// TripletMarginLoss_56676388438753
// MI455X (gfx1250) — compile-verified
//
#include <hip/hip_runtime.h>

typedef __attribute__((ext_vector_type(2))) float v2f;
typedef __attribute__((ext_vector_type(8))) float v8f;

namespace {
constexpr int kN = 512;   // number of embeddings
constexpr int kD = 128;   // embedding dim
constexpr float kMargin = 1.0f;
constexpr float kEps = 1e-8f;
}

// ---------------------------------------------------------------------------
// Kernel 1: dm = E * E^T using native fp32 WMMA (V_WMMA_F32_16X16X4_F32).
// One wave32 computes one 16x16 tile of dm. 32 K-steps of K=4 cover D=128.
//
// A-tile (16x4 f32, 2 VGPRs): lane L holds E[i0 + L%16][kk + 2*(L/16) + {0,1}]
// B-tile (4x16 f32, 2 VGPRs): since B = (E j-tile)^T, lane L holds
//                             E[j0 + L%16][kk + 2*(L/16) + {0,1}]  (same gather)
// C/D (16x16 f32, 8 VGPRs):   VGPR r -> row = r + 8*(L/16), col = L%16
// ---------------------------------------------------------------------------
__global__ __launch_bounds__(256) void gemm_ee_t_wmma(const float* __restrict__ E,
                                                      float* __restrict__ dm) {
  const int wave = threadIdx.x >> 5;
  const int lane = threadIdx.x & 31;
  const int tile = blockIdx.x * 8 + wave;      // 1024 tiles total
  const int ti = tile >> 5;                    // tile row   (i block)
  const int tj = tile & 31;                    // tile col   (j block)
  const int m  = lane & 15;                    // row-in-tile (A) / col-in-tile (B)
  const int kh = (lane >> 4) << 1;             // 0 for lanes 0-15, 2 for lanes 16-31

  const float* arow = E + (size_t)(ti * 16 + m) * kD + kh;
  const float* brow = E + (size_t)(tj * 16 + m) * kD + kh;

  v8f acc = {};
#pragma unroll 4
  for (int kk = 0; kk < kD; kk += 4) {
    v2f a = *(const v2f*)(arow + kk);          // 8-byte aligned float2 load
    v2f b = *(const v2f*)(brow + kk);
    // 8 args: (neg_a, A, neg_b, B, c_mod, C, reuse_a, reuse_b); neg must be 0 for f32
    acc = __builtin_amdgcn_wmma_f32_16x16x4_f32(false, a, false, b,
                                                (short)0, acc, false, false);
  }

  const int col  = tj * 16 + m;
  const int row0 = ti * 16 + ((lane >> 4) << 3);
#pragma unroll
  for (int r = 0; r < 8; ++r)
    dm[(size_t)(row0 + r) * kN + col] = acc[r];
}

// ---------------------------------------------------------------------------
// Kernel 2: per-anchor reduction. Block i handles anchor i.
// sum_i = sum_{j in P_i, k in N_i} relu(dm[i,j] - dm[i,k] + margin)
// cnt_i = #{ (j,k) : dm[i,j] - dm[i,k] + margin > EPS }
// Deterministic: positives list built serially; fixed-order LDS tree reduce.
// ---------------------------------------------------------------------------
__global__ __launch_bounds__(256) void triplet_anchor_reduce(
    const float* __restrict__ dm, const int* __restrict__ labels,
    float* __restrict__ psum, float* __restrict__ pcnt) {
  __shared__ float row[kN];
  __shared__ int   lab[kN];
  __shared__ short poslist[kN];
  __shared__ int   np_s;
  __shared__ float rs[256];
  __shared__ float rc[256];

  const int i = blockIdx.x;
  const int t = threadIdx.x;

  for (int x = t; x < kN; x += 256) {
    row[x] = dm[(size_t)i * kN + x];
    lab[x] = labels[x];
  }
  __syncthreads();

  const int li = lab[i];
  if (t == 0) {                       // deterministic positive-list build
    int np = 0;
    for (int j = 0; j < kN; ++j)
      if (j != i && lab[j] == li) poslist[np++] = (short)j;
    np_s = np;
  }
  __syncthreads();

  const int np = np_s;
  float sum = 0.0f, cnt = 0.0f;
  for (int p = 0; p < np; ++p) {      // ~7 positives per anchor
    const float dij = row[poslist[p]] + kMargin;
    for (int k = t; k < kN; k += 256) {
      if (lab[k] != li) {             // negative => k != i, k != j implied
        const float v = dij - row[k];
        sum += fmaxf(v, 0.0f);
        cnt += (v > kEps) ? 1.0f : 0.0f;
      }
    }
  }

  rs[t] = sum; rc[t] = cnt;
  __syncthreads();
  for (int s = 128; s > 0; s >>= 1) {
    if (t < s) { rs[t] += rs[t + s]; rc[t] += rc[t + s]; }
    __syncthreads();
  }
  if (t == 0) { psum[i] = rs[0]; pcnt[i] = rc[0]; }
}

// ---------------------------------------------------------------------------
// Kernel 3: reduce 512 per-anchor partials (fixed order) -> scalar loss.
// ---------------------------------------------------------------------------
__global__ __launch_bounds__(256) void triplet_finalize(
    const float* __restrict__ psum, const float* __restrict__ pcnt,
    float* __restrict__ out) {
  __shared__ float rs[256];
  __shared__ float rc[256];
  const int t = threadIdx.x;
  rs[t] = psum[t] + psum[t + 256];
  rc[t] = pcnt[t] + pcnt[t + 256];
  __syncthreads();
  for (int s = 128; s > 0; s >>= 1) {
    if (t < s) { rs[t] += rs[t + s]; rc[t] += rc[t + s]; }
    __syncthreads();
  }
  if (t == 0) out[0] = rs[0] / (rc[0] + kEps);
}

extern "C" void kernel_launch(void* const* d_in, const int* in_sizes, int n_in,
                              void* d_out, int out_size, void* d_ws, size_t ws_size,
                              hipStream_t stream) {
  (void)in_sizes; (void)n_in; (void)out_size; (void)ws_size;
  const float* E      = (const float*)d_in[0];
  const int*   labels = (const int*)d_in[1];

  float* dm    = (float*)d_ws;              // 512*512 f32 = 1 MB
  float* psum  = dm + (size_t)kN * kN;      // 512 f32
  float* pcnt  = psum + kN;                 // 512 f32
  float* out   = (float*)d_out;

  // 1024 tiles, 8 waves (one tile each) per 256-thread block -> 128 blocks
  gemm_ee_t_wmma<<<128, 256, 0, stream>>>(E, dm);
  triplet_anchor_reduce<<<kN, 256, 0, stream>>>(dm, labels, psum, pcnt);
  triplet_finalize<<<1, 256, 0, stream>>>(psum, pcnt, out);
}